// TopicAttention_39427799777824
// MI455X (gfx1250) — compile-verified
//
#include <hip/hip_runtime.h>
#include <hip/hip_bf16.h>
#include <math.h>

// Problem constants
#define BB   128
#define TT   512
#define II   300
#define HH   150
#define H3   450
#define KK   6
#define CC   300
#define DD   300   // 2H
#define THH  20
#define CLS  5

typedef float v2f __attribute__((ext_vector_type(2)));
typedef float v8f __attribute__((ext_vector_type(8)));

__device__ __forceinline__ float sigmoidf(float x) {
    return 1.0f / (1.0f + __expf(-x));
}

// ---------------------------------------------------------------------------
// Kernel 1: gi = x @ Wih^T + bih for both directions, stored as gi[t][b][n].
//
// Block tile: 128 (M) x 32 (N), K chunked at 32.  A/B chunks are staged in
// LDS with double-buffered GLOBAL_LOAD_ASYNC_TO_LDS_B128 (ASYNCcnt) so the
// fp32 WMMA chain overlaps the global->LDS DMA.  8 waves = 8 M-subtiles,
// each wave holds two 16x16 fp32 accumulators (N-subtiles) and reuses its
// A fragment for both -> 2 v_wmma per A ds_load.
// K = 300 = 9*32 + 12; the last chunk is exactly 3 K4 steps (12 k's), so the
// tail needs no zero-fill and EXEC stays full for all WMMAs.
// ---------------------------------------------------------------------------
#define MB    128
#define NBW   32
#define KCH   32
#define ASTR  36                 // LDS row stride (floats): 144B, 16B-aligned
#define NBLK  15                 // ceil(450/32)
#define MBLK  512                // 65536/128
#define NCHUNK 10                // ceil(300/32)

__global__ void __launch_bounds__(256)
gi_gemm_kernel(const float* __restrict__ x,
               const float* __restrict__ Wih_f,
               const float* __restrict__ bih_f,
               const float* __restrict__ Wih_b,
               const float* __restrict__ bih_b,
               float* __restrict__ gi_f,
               float* __restrict__ gi_b) {
    __shared__ __align__(16) float Abuf[2][MB * ASTR];   // 2 x 18 KB
    __shared__ __align__(16) float Bbuf[2][NBW * ASTR];  // 2 x 4.5 KB

    const int tid  = threadIdx.x;
    const int lane = tid & 31;
    const int wave = tid >> 5;           // == M-subtile index (0..7)

    int blk = blockIdx.x;
    const int dir = blk / (MBLK * NBLK);
    blk -= dir * (MBLK * NBLK);
    const int mb = blk / NBLK;
    const int nb = blk - mb * NBLK;

    const float* __restrict__ W   = dir ? Wih_b : Wih_f;
    const float* __restrict__ bih = dir ? bih_b : bih_f;
    float*       __restrict__ gi  = dir ? gi_b  : gi_f;

    const float* __restrict__ xblk = x + (long)mb * MB * II;

    // Issue one K-chunk of async global->LDS copies.  Exactly 5
    // global_load_async_to_lds_b128 per thread position, every chunk
    // (addresses clamped instead of lanes disabled, so the per-wave
    // ASYNCcnt increment is a compile-time constant and s_wait_asynccnt
    // retires whole chunks).
    auto issue_chunk = [&](int c, int buf) {
        const int kbase = c * KCH;
        const int ngrp  = (c == NCHUNK - 1) ? 3 : 8;   // 16B groups per row
        // A: 128 rows x ngrp groups (1024 lanes-worth -> 4 instructions)
        #pragma unroll
        for (int i = 0; i < 4; ++i) {
            const int g    = tid + i * 256;
            const int row  = g >> 3;
            int       grp  = g & 7;
            if (grp >= ngrp) grp = ngrp - 1;           // clamp: dup, not skip
            const unsigned lds =
                (unsigned)(size_t)&Abuf[buf][row * ASTR + grp * 4];
            const float* src = xblk + (long)row * II + kbase + grp * 4;
            asm volatile("global_load_async_to_lds_b128 %0, %1, off"
                         :: "v"(lds), "v"(src) : "memory");
        }
        // B: 32 rows x ngrp groups (256 lanes-worth -> 1 instruction)
        {
            const int row = tid >> 3;
            int       grp = tid & 7;
            if (grp >= ngrp) grp = ngrp - 1;
            int wr = nb * NBW + row;
            if (wr > H3 - 1) wr = H3 - 1;              // clamp OOB N rows
            const unsigned lds =
                (unsigned)(size_t)&Bbuf[buf][row * ASTR + grp * 4];
            const float* src = W + (long)wr * II + kbase + grp * 4;
            asm volatile("global_load_async_to_lds_b128 %0, %1, off"
                         :: "v"(lds), "v"(src) : "memory");
        }
    };

    v8f acc0 = {};
    v8f acc1 = {};
    const int khalf = (lane >> 4) * 2;

    issue_chunk(0, 0);

    for (int c = 0; c < NCHUNK; ++c) {
        const int buf = c & 1;
        if (c < NCHUNK - 1) {
            issue_chunk(c + 1, buf ^ 1);
            asm volatile("s_wait_asynccnt 0x5" ::: "memory"); // chunk c landed
        } else {
            asm volatile("s_wait_asynccnt 0x0" ::: "memory");
        }
        __syncthreads();                // everyone's chunk-c data visible

        const float* __restrict__ arow =
            &Abuf[buf][(wave * 16 + (lane & 15)) * ASTR];
        const float* __restrict__ brow0 = &Bbuf[buf][(lane & 15) * ASTR];
        const float* __restrict__ brow1 = &Bbuf[buf][(16 + (lane & 15)) * ASTR];

        const int ksteps = (c == NCHUNK - 1) ? 3 : 8;
        for (int kk = 0; kk < ksteps; ++kk) {
            const int k0 = kk * 4 + khalf;
            v2f a, b0, b1;
            a.x  = arow[k0];   a.y  = arow[k0 + 1];
            b0.x = brow0[k0];  b0.y = brow0[k0 + 1];
            b1.x = brow1[k0];  b1.y = brow1[k0 + 1];
            acc0 = __builtin_amdgcn_wmma_f32_16x16x4_f32(
                       false, a, false, b0, (short)0, acc0, false, false);
            acc1 = __builtin_amdgcn_wmma_f32_16x16x4_f32(
                       false, a, false, b1, (short)0, acc1, false, false);
        }
        __syncthreads();                // LDS reads done before buf reuse
    }

    // Store: C VGPR r -> rows {r | lanes<16 ; r+8 | lanes>=16}, col = lane&15
    #pragma unroll
    for (int nt = 0; nt < 2; ++nt) {
        const v8f acc = nt ? acc1 : acc0;
        const int n = nb * NBW + nt * 16 + (lane & 15);
        if (n < H3) {
            const float bias  = bih[n];
            const int   mbase = mb * MB + wave * 16 + (lane >> 4) * 8;
            #pragma unroll
            for (int r = 0; r < 8; ++r) {
                const int m  = mbase + r;
                const int bi = m >> 9;          // m / T
                const int ti = m & 511;         // m % T
                gi[((long)ti * BB + bi) * H3 + n] = acc[r] + bias;
            }
        }
    }
}

// ---------------------------------------------------------------------------
// Kernel 2: persistent GRU scan. grid = 2 (dir), block = 1024 (32 waves).
// LDS: h[128][152] (77.8 KB) + gh[128][456] (233.5 KB) = 311 KB / WGP.
// Per step: gh = h @ Whh^T via fp32 WMMA (232 tiles over 32 waves), then
// elementwise gate math, h rewritten in LDS, output row written to seq.
// gi[t+1] (230 KB) is prefetched (global_prefetch_b8) during the WMMA phase.
// ---------------------------------------------------------------------------
#define HSTR 152
#define GSTR 456
#define NT_T 29                  // ceil(450/16)

__global__ void gru_scan_kernel(const float* __restrict__ gi_f,
                                const float* __restrict__ gi_b,
                                const float* __restrict__ Whh_f,
                                const float* __restrict__ bhh_f,
                                const float* __restrict__ Whh_b,
                                const float* __restrict__ bhh_b,
                                float* __restrict__ seq,
                                float* __restrict__ hidden) {
    extern __shared__ float smem[];
    float* hbuf  = smem;                 // 128 x HSTR
    float* ghbuf = smem + BB * HSTR;     // 128 x GSTR

    const int dir  = blockIdx.x;
    const int tid  = threadIdx.x;
    const int lane = tid & 31;
    const int wave = tid >> 5;

    const float* __restrict__ gi  = dir ? gi_b  : gi_f;
    const float* __restrict__ Whh = dir ? Whh_b : Whh_f;
    const float* __restrict__ bhh = dir ? bhh_b : bhh_f;

    for (int i = tid; i < BB * HSTR; i += 1024) hbuf[i] = 0.0f;
    __syncthreads();

    for (int s = 0; s < TT; ++s) {
        const int t = dir ? (TT - 1 - s) : s;

        // prefetch next step's gi slab into cache while WMMAs run
        if (s + 1 < TT) {
            const int t2 = dir ? (TT - 2 - s) : (s + 1);
            const float* __restrict__ nxt = gi + ((long)t2 * BB) * H3;
            for (int i = tid; i < (BB * H3) / 32; i += 1024)
                __builtin_prefetch(nxt + (long)i * 32, 0, 0);
        }

        // ---- phase 1: gh = h @ Whh^T, fp32 WMMA, A (h) from LDS, B (Whh) from global ----
        for (int tt = wave; tt < 8 * NT_T; tt += 32) {
            const int mt  = tt / NT_T;
            const int nt  = tt - mt * NT_T;
            const int nb0 = nt * 16 + (lane & 15);
            const int nbc = (nb0 < H3) ? nb0 : (H3 - 1);
            const int khalf = (lane >> 4) * 2;
            const float* __restrict__ hrow = hbuf + (mt * 16 + (lane & 15)) * HSTR;
            const float* __restrict__ wrow = Whh + (long)nbc * HH; // B[k][n]=Whh[n*HH+k]

            v8f acc = {};
            #pragma unroll 2
            for (int kk = 0; kk < 38; ++kk) {       // K=150, tail zero-filled
                const int k0 = kk * 4 + khalf;
                v2f a, b;
                a.x = (k0     < HH) ? hrow[k0]     : 0.0f;
                a.y = (k0 + 1 < HH) ? hrow[k0 + 1] : 0.0f;
                b.x = (k0     < HH) ? wrow[k0]     : 0.0f;
                b.y = (k0 + 1 < HH) ? wrow[k0 + 1] : 0.0f;
                acc = __builtin_amdgcn_wmma_f32_16x16x4_f32(
                          false, a, false, b, (short)0, acc, false, false);
            }
            if (nb0 < H3) {
                const int mbase = mt * 16 + (lane >> 4) * 8;
                #pragma unroll
                for (int r = 0; r < 8; ++r)
                    ghbuf[(mbase + r) * GSTR + nb0] = acc[r];
            }
        }
        __syncthreads();

        // ---- phase 2: gate math + h update + seq output ----
        const long gibase = ((long)t * BB) * H3;
        for (int e = tid; e < BB * HH; e += 1024) {
            const int bi = e / HH;
            const int j  = e - bi * HH;
            const float hval = hbuf[bi * HSTR + j];
            const float ghr  = ghbuf[bi * GSTR + j]          + bhh[j];
            const float ghz  = ghbuf[bi * GSTR + HH + j]     + bhh[HH + j];
            const float ghn  = ghbuf[bi * GSTR + 2 * HH + j] + bhh[2 * HH + j];
            const long  gidx = gibase + (long)bi * H3;
            const float gir  = gi[gidx + j];
            const float giz  = gi[gidx + HH + j];
            const float gin  = gi[gidx + 2 * HH + j];
            const float r = sigmoidf(gir + ghr);
            const float z = sigmoidf(giz + ghz);
            const float n = tanhf(gin + r * ghn);
            const float hn = (1.0f - z) * n + z * hval;
            hbuf[bi * HSTR + j] = hn;
            seq[((long)bi * TT + t) * DD + dir * HH + j] = hn;
        }
        __syncthreads();
    }

    // final hidden state
    for (int e = tid; e < BB * HH; e += 1024) {
        const int bi = e / HH;
        const int j  = e - bi * HH;
        hidden[bi * DD + dir * HH + j] = hbuf[bi * HSTR + j];
    }
}

// ---------------------------------------------------------------------------
// Kernel 3: context[b,k,d] = tanh( [attn_ctx[k], hidden[b]] . Wattn[k,:,d] + battn[k,d] )
// ---------------------------------------------------------------------------
__global__ void context_kernel(const float* __restrict__ attn_context,
                               const float* __restrict__ hidden,
                               const float* __restrict__ Wattn,
                               const float* __restrict__ battn,
                               float* __restrict__ context) {
    const int b = blockIdx.x / KK;
    const int k = blockIdx.x - b * KK;
    __shared__ float vin[CC + DD];
    for (int i = threadIdx.x; i < CC + DD; i += blockDim.x)
        vin[i] = (i < CC) ? attn_context[k * CC + i] : hidden[b * DD + (i - CC)];
    __syncthreads();
    for (int d = threadIdx.x; d < DD; d += blockDim.x) {
        float s = battn[k * DD + d];
        const float* __restrict__ w = Wattn + ((long)k * (CC + DD)) * DD + d;
        for (int c = 0; c < CC + DD; ++c) s += vin[c] * w[(long)c * DD];
        context[((long)b * KK + k) * DD + d] = tanhf(s);
    }
}

// ---------------------------------------------------------------------------
// Kernel 4a: zero the regularizer accumulator.
// Kernel 4b: per-batch Gram-matrix orthogonality penalty, mean via atomicAdd.
// ---------------------------------------------------------------------------
__global__ void zero_reg_kernel(float* __restrict__ p) {
    if (blockIdx.x == 0 && threadIdx.x == 0) p[0] = 0.0f;
}

__global__ void gram_kernel(const float* __restrict__ context,
                            float* __restrict__ reg_out) {
    const int b   = blockIdx.x;
    const int tid = threadIdx.x;
    __shared__ float G[KK * KK];
    __shared__ float nrm[KK];
    if (tid < KK * KK) {
        const int k = tid / KK, j = tid - k * KK;
        const float* __restrict__ ck = context + ((long)b * KK + k) * DD;
        const float* __restrict__ cj = context + ((long)b * KK + j) * DD;
        float s = 0.0f;
        for (int d = 0; d < DD; ++d) s += ck[d] * cj[d];
        G[tid] = s;
    }
    __syncthreads();
    if (tid < KK) {
        const float v = sqrtf(G[tid * (KK + 1)]);
        nrm[tid] = (v > 1e-12f) ? v : 1e-12f;
    }
    __syncthreads();
    if (tid == 0) {
        float acc = 0.0f;
        for (int k = 0; k < KK; ++k)
            for (int j = 0; j < KK; ++j) {
                const float g = G[k * KK + j] / (nrm[k] * nrm[j]);
                const float d = g - (k == j ? 1.0f : 0.0f);
                acc += d * d;
            }
        atomicAdd(reg_out, sqrtf(acc) * (1.0f / (float)BB));
    }
}

// ---------------------------------------------------------------------------
// Kernel 5: per-batch energy -> softmax(axis=t) -> pooled -> topic -> logits
// -> softmax.  One block per batch element.
// ---------------------------------------------------------------------------
__global__ void pool_head_kernel(const float* __restrict__ seq,
                                 const float* __restrict__ context,
                                 const float* __restrict__ Wtop,
                                 const float* __restrict__ btop,
                                 const float* __restrict__ Wout,
                                 const float* __restrict__ bout,
                                 float* __restrict__ out) {
    const int b = blockIdx.x;
    __shared__ float en[TT * KK];         // energy -> probs (12 KB)
    __shared__ float pooled[KK * DD];     // 7.2 KB
    __shared__ float feats[KK * THH];
    __shared__ float logits[CLS];
    const float* __restrict__ seqb = seq + (long)b * TT * DD;
    const float* __restrict__ ctxb = context + (long)b * KK * DD;

    for (int t = threadIdx.x; t < TT; t += blockDim.x) {
        const float* __restrict__ srow = seqb + (long)t * DD;
        float s[KK] = {0, 0, 0, 0, 0, 0};
        for (int d = 0; d < DD; ++d) {
            const float sv = srow[d];
            #pragma unroll
            for (int k = 0; k < KK; ++k) s[k] += sv * ctxb[k * DD + d];
        }
        #pragma unroll
        for (int k = 0; k < KK; ++k) en[t * KK + k] = s[k];
    }
    __syncthreads();

    if (threadIdx.x < KK) {               // softmax over t per topic k
        const int k = threadIdx.x;
        float mx = -INFINITY;
        for (int t = 0; t < TT; ++t) mx = fmaxf(mx, en[t * KK + k]);
        float sum = 0.0f;
        for (int t = 0; t < TT; ++t) {
            const float e = __expf(en[t * KK + k] - mx);
            en[t * KK + k] = e;
            sum += e;
        }
        const float inv = 1.0f / sum;
        for (int t = 0; t < TT; ++t) en[t * KK + k] *= inv;
    }
    __syncthreads();

    for (int idx = threadIdx.x; idx < KK * DD; idx += blockDim.x) {
        const int k = idx / DD, d = idx - k * DD;
        float s = 0.0f;
        for (int t = 0; t < TT; ++t) s += seqb[(long)t * DD + d] * en[t * KK + k];
        pooled[k * DD + d] = s;
    }
    __syncthreads();

    for (int idx = threadIdx.x; idx < KK * THH; idx += blockDim.x) {
        const int k = idx / THH, h2 = idx - k * THH;
        float s = btop[k * THH + h2];
        const float* __restrict__ pk = pooled + k * DD;
        const float* __restrict__ w  = Wtop + ((long)k * DD) * THH + h2;
        for (int d = 0; d < DD; ++d) s += pk[d] * w[(long)d * THH];
        feats[idx] = fmaxf(s, 0.0f);
    }
    __syncthreads();

    if (threadIdx.x < CLS) {
        const int c = threadIdx.x;
        float s = bout[c];
        for (int f = 0; f < KK * THH; ++f) s += feats[f] * Wout[f * CLS + c];
        logits[c] = s;
    }
    __syncthreads();

    if (threadIdx.x == 0) {
        float mx = logits[0];
        for (int c = 1; c < CLS; ++c) mx = fmaxf(mx, logits[c]);
        float e[CLS], sum = 0.0f;
        for (int c = 0; c < CLS; ++c) { e[c] = __expf(logits[c] - mx); sum += e[c]; }
        const float inv = 1.0f / sum;
        for (int c = 0; c < CLS; ++c) out[b * CLS + c] = e[c] * inv;
    }
}

// ---------------------------------------------------------------------------
extern "C" void kernel_launch(void* const* d_in, const int* in_sizes, int n_in,
                              void* d_out, int out_size, void* d_ws, size_t ws_size,
                              hipStream_t stream) {
    const float* x            = (const float*)d_in[0];
    const float* Wih_f        = (const float*)d_in[1];
    const float* Whh_f        = (const float*)d_in[2];
    const float* bih_f        = (const float*)d_in[3];
    const float* bhh_f        = (const float*)d_in[4];
    const float* Wih_b        = (const float*)d_in[5];
    const float* Whh_b        = (const float*)d_in[6];
    const float* bih_b        = (const float*)d_in[7];
    const float* bhh_b        = (const float*)d_in[8];
    const float* attn_context = (const float*)d_in[9];
    const float* Wattn        = (const float*)d_in[10];
    const float* battn        = (const float*)d_in[11];
    const float* Wtop         = (const float*)d_in[12];
    const float* btop         = (const float*)d_in[13];
    const float* Wout         = (const float*)d_in[14];
    const float* bout         = (const float*)d_in[15];

    float* outp = (float*)d_out;          // [B*CLS | reg] = 641 floats

    // workspace layout (floats)
    float* ws      = (float*)d_ws;
    float* gi_f    = ws;                                   // 65536*450
    float* gi_b    = gi_f + (long)BB * TT * H3;
    float* seq     = gi_b + (long)BB * TT * H3;            // B*T*2H
    float* hidden  = seq + (long)BB * TT * DD;             // B*2H
    float* context = hidden + (long)BB * DD;               // B*K*2H

    // 1) input projections, async-LDS double-buffered fp32 WMMA GEMM
    gi_gemm_kernel<<<2 * MBLK * NBLK, 256, 0, stream>>>(
        x, Wih_f, bih_f, Wih_b, bih_b, gi_f, gi_b);

    // 2) persistent GRU scan, one 1024-thread workgroup per direction
    const size_t shmem = (size_t)(BB * HSTR + BB * GSTR) * sizeof(float); // 311296 B
    gru_scan_kernel<<<2, 1024, shmem, stream>>>(
        gi_f, gi_b, Whh_f, bhh_f, Whh_b, bhh_b, seq, hidden);

    // 3) attention context
    context_kernel<<<BB * KK, 128, 0, stream>>>(attn_context, hidden, Wattn, battn, context);

    // 4) Gram regularizer (mean over batch)
    zero_reg_kernel<<<1, 32, 0, stream>>>(outp + BB * CLS);
    gram_kernel<<<BB, 64, 0, stream>>>(context, outp + BB * CLS);

    // 5) energy/softmax/pool/topic/logits
    pool_head_kernel<<<BB, 256, 0, stream>>>(seq, context, Wtop, btop, Wout, bout, outp);
}